// Base_55499567399232
// MI455X (gfx1250) — compile-verified
//
#include <hip/hip_runtime.h>
#include <math.h>

#define NDIM 64
#define SCALEF 0.125f  // 1/sqrt(64)

typedef float v2f __attribute__((ext_vector_type(2)));
typedef float v8f __attribute__((ext_vector_type(8)));

// ---------------------------------------------------------------- helpers
__device__ __forceinline__ void atomicMaxFloat(float* addr, float val) {
    // Monotonic-bit-pattern trick; addr initialized to -INFINITY.
    if (val >= 0.0f) atomicMax((int*)addr, __float_as_int(val));
    else             atomicMin((unsigned int*)addr, __float_as_uint(val));
}

// ---------------------------------------------------------------- kernels
__global__ void k_gather(const int* __restrict__ atoms,
                         const float* __restrict__ emb,
                         float* __restrict__ x, int N) {
    int tid = blockIdx.x * blockDim.x + threadIdx.x;
    if (tid >= N * NDIM) return;
    int node = tid >> 6, d = tid & 63;
    x[tid] = emb[atoms[node] * NDIM + d];
}

__global__ void k_init(float* __restrict__ agg_r, float* __restrict__ agg_c,
                       float* __restrict__ m_r, float* __restrict__ m_c,
                       float* __restrict__ s_r, float* __restrict__ s_c, int N) {
    int tid = blockIdx.x * blockDim.x + threadIdx.x;
    if (tid < N * NDIM) { agg_r[tid] = 0.0f; agg_c[tid] = 0.0f; }
    if (tid < N) {
        m_r[tid] = -INFINITY; m_c[tid] = -INFINITY;
        s_r[tid] = 0.0f;      s_c[tid] = 0.0f;
    }
}

// 6 projections: P[j] = x @ W[j]; one 16x16 output tile per wave, K=64.
// grid = (N/16, 6), block = 128 (4 waves, one per 16-col tile).
__global__ void k_proj6(const float* __restrict__ x,
                        const float* __restrict__ w0, const float* __restrict__ w1,
                        const float* __restrict__ w2, const float* __restrict__ w3,
                        const float* __restrict__ w4, const float* __restrict__ w5,
                        float* __restrict__ P, int N) {
    const float* W;
    switch (blockIdx.y) {
        case 0: W = w0; break; case 1: W = w1; break; case 2: W = w2; break;
        case 3: W = w3; break; case 4: W = w4; break; default: W = w5; break;
    }
    float* out = P + (size_t)blockIdx.y * (size_t)N * NDIM;

    int wave = threadIdx.x >> 5;
    int lane = threadIdx.x & 31;
    int row0 = blockIdx.x * 16;
    int n0   = wave * 16;
    int m     = lane & 15;
    int khalf = (lane >> 4) * 2;          // 0 for lanes 0-15, 2 for lanes 16-31
    int n     = n0 + m;

    int arow_idx = row0 + m; if (arow_idx >= N) arow_idx = N - 1;
    const float* arow = x + (size_t)arow_idx * NDIM;

    v8f c = {};
    #pragma unroll
    for (int k0 = 0; k0 < NDIM; k0 += 4) {
        int kb = k0 + khalf;
        v2f a = *(const v2f*)(arow + kb);                       // A[m][kb], A[m][kb+1]
        v2f b; b.x = W[kb * NDIM + n]; b.y = W[(kb + 1) * NDIM + n];  // B col-major per lane
        c = __builtin_amdgcn_wmma_f32_16x16x4_f32(false, a, false, b,
                                                  (short)0, c, false, false);
    }
    int col    = n0 + (lane & 15);
    int rowadd = (lane >> 4) * 8;
    #pragma unroll
    for (int r = 0; r < 8; ++r) {
        int row = row0 + r + rowadd;
        if (row < N) out[(size_t)row * NDIM + col] = c[r];
    }
}

// Pass 1: per-edge scores for both directions; 4 lanes per edge.
__global__ void k_edge_scores(const int* __restrict__ src, const int* __restrict__ dst,
                              const int* __restrict__ eid,
                              const float* __restrict__ Qr, const float* __restrict__ Kr,
                              const float* __restrict__ Qc, const float* __restrict__ Kc,
                              const float* __restrict__ Eer, const float* __restrict__ Eec,
                              float* __restrict__ scr, float* __restrict__ scc,
                              float* __restrict__ m_r, float* __restrict__ m_c, int E) {
    int tid = blockIdx.x * blockDim.x + threadIdx.x;
    int e = tid >> 2;
    if (e >= E) return;
    int p = tid & 3;
    int s = src[e], d = dst[e], t = eid[e];

    const float4* qrp = (const float4*)(Qr + (size_t)d * NDIM);
    const float4* krp = (const float4*)(Kr + (size_t)s * NDIM);
    const float4* erp = (const float4*)(Eer + (size_t)t * NDIM);
    const float4* qcp = (const float4*)(Qc + (size_t)s * NDIM);
    const float4* kcp = (const float4*)(Kc + (size_t)d * NDIM);
    const float4* ecp = (const float4*)(Eec + (size_t)t * NDIM);

    float accr = 0.0f, accc = 0.0f;
    #pragma unroll
    for (int i = 0; i < 4; ++i) {
        int ci = p * 4 + i;
        float4 q = qrp[ci], k = krp[ci], ee = erp[ci];
        accr += q.x * (k.x + ee.x) + q.y * (k.y + ee.y)
              + q.z * (k.z + ee.z) + q.w * (k.w + ee.w);
        float4 q2 = qcp[ci], k2 = kcp[ci], e2 = ecp[ci];
        accc += q2.x * (k2.x + e2.x) + q2.y * (k2.y + e2.y)
              + q2.z * (k2.z + e2.z) + q2.w * (k2.w + e2.w);
    }
    accr += __shfl_xor(accr, 1, 32); accr += __shfl_xor(accr, 2, 32);
    accc += __shfl_xor(accc, 1, 32); accc += __shfl_xor(accc, 2, 32);

    if (p == 0) {
        float sr = accr * SCALEF, sc = accc * SCALEF;
        scr[e] = sr; scc[e] = sc;
        atomicMaxFloat(&m_r[d], sr);
        atomicMaxFloat(&m_c[s], sc);
    }
}

// Pass 2: exp(score - max) and segment sums.
__global__ void k_edge_exp(const int* __restrict__ src, const int* __restrict__ dst,
                           float* __restrict__ scr, float* __restrict__ scc,
                           const float* __restrict__ m_r, const float* __restrict__ m_c,
                           float* __restrict__ s_r, float* __restrict__ s_c, int E) {
    int e = blockIdx.x * blockDim.x + threadIdx.x;
    if (e >= E) return;
    int s = src[e], d = dst[e];
    float exr = expf(scr[e] - m_r[d]);
    float exc = expf(scc[e] - m_c[s]);
    scr[e] = exr; scc[e] = exc;
    atomicAdd(&s_r[d], exr);
    atomicAdd(&s_c[s], exc);
}

// Pass 3: normalized attention * V scatter-add; 16 lanes per edge, 4 floats each.
__global__ void k_edge_aggr(const int* __restrict__ src, const int* __restrict__ dst,
                            const float* __restrict__ ex_r, const float* __restrict__ ex_c,
                            const float* __restrict__ s_r, const float* __restrict__ s_c,
                            const float* __restrict__ Vr, const float* __restrict__ Vc,
                            float* __restrict__ agg_r, float* __restrict__ agg_c, int E) {
    long long tid = (long long)blockIdx.x * blockDim.x + threadIdx.x;
    int e = (int)(tid >> 4);
    if (e >= E) return;
    int p = (int)(tid & 15);
    int s = src[e], d = dst[e];
    float wr = ex_r[e] / (s_r[d] + 1e-16f);
    float wc = ex_c[e] / (s_c[s] + 1e-16f);
    float4 vr = ((const float4*)(Vr + (size_t)s * NDIM))[p];
    float4 vc = ((const float4*)(Vc + (size_t)d * NDIM))[p];
    float* orp = agg_r + (size_t)d * NDIM + p * 4;
    float* ocp = agg_c + (size_t)s * NDIM + p * 4;
    atomicAdd(orp + 0, wr * vr.x); atomicAdd(orp + 1, wr * vr.y);
    atomicAdd(orp + 2, wr * vr.z); atomicAdd(orp + 3, wr * vr.w);
    atomicAdd(ocp + 0, wc * vc.x); atomicAdd(ocp + 1, wc * vc.y);
    atomicAdd(ocp + 2, wc * vc.z); atomicAdd(ocp + 3, wc * vc.w);
}

// Final: xout = gelu(concat(agg_r + x, agg_c) @ Wa + ba), K=128 WMMA GEMM.
__global__ void k_out(const float* __restrict__ x,
                      const float* __restrict__ agg_r, const float* __restrict__ agg_c,
                      const float* __restrict__ Wa, const float* __restrict__ ba,
                      float* __restrict__ xout, int N) {
    int wave = threadIdx.x >> 5;
    int lane = threadIdx.x & 31;
    int row0 = blockIdx.x * 16;
    int n0   = wave * 16;
    int m     = lane & 15;
    int khalf = (lane >> 4) * 2;
    int n     = n0 + m;

    int arow_idx = row0 + m; if (arow_idx >= N) arow_idx = N - 1;
    size_t rowoff = (size_t)arow_idx * NDIM;

    v8f c = {};
    #pragma unroll
    for (int k0 = 0; k0 < 2 * NDIM; k0 += 4) {
        int kb = k0 + khalf;
        v2f a;
        if (k0 < NDIM) {  // wave-uniform branch (kb in {k0, k0+2}, boundary aligned)
            v2f ar = *(const v2f*)(agg_r + rowoff + kb);
            v2f xr = *(const v2f*)(x + rowoff + kb);
            a.x = ar.x + xr.x; a.y = ar.y + xr.y;
        } else {
            a = *(const v2f*)(agg_c + rowoff + (kb - NDIM));
        }
        v2f b; b.x = Wa[kb * NDIM + n]; b.y = Wa[(kb + 1) * NDIM + n];
        c = __builtin_amdgcn_wmma_f32_16x16x4_f32(false, a, false, b,
                                                  (short)0, c, false, false);
    }
    int col    = n0 + (lane & 15);
    int rowadd = (lane >> 4) * 8;
    float bv = ba[col];
    #pragma unroll
    for (int r = 0; r < 8; ++r) {
        int row = row0 + r + rowadd;
        if (row < N) {
            float v = c[r] + bv;
            float g = 0.5f * v * (1.0f + erff(v * 0.70710678118654752f));
            xout[(size_t)row * NDIM + col] = g;
        }
    }
}

// ---------------------------------------------------------------- launcher
extern "C" void kernel_launch(void* const* d_in, const int* in_sizes, int n_in,
                              void* d_out, int out_size, void* d_ws, size_t ws_size,
                              hipStream_t stream) {
    const int*   atoms = (const int*)d_in[0];
    const int*   ei    = (const int*)d_in[1];
    const int*   eids  = (const int*)d_in[2];
    const float* emb   = (const float*)d_in[3];
    const float* Wq_r  = (const float*)d_in[4];
    const float* Wk_r  = (const float*)d_in[5];
    const float* Wv_r  = (const float*)d_in[6];
    const float* Ee_r  = (const float*)d_in[7];
    const float* Wq_c  = (const float*)d_in[8];
    const float* Wk_c  = (const float*)d_in[9];
    const float* Wv_c  = (const float*)d_in[10];
    const float* Ee_c  = (const float*)d_in[11];
    const float* Wa    = (const float*)d_in[12];
    const float* ba    = (const float*)d_in[13];

    const int N = in_sizes[0];
    const int E = in_sizes[2];
    const int L = in_sizes[4] / (NDIM * NDIM);
    const int T = in_sizes[7] / (L * NDIM);

    const int* srcp = ei;       // edge_index[0]
    const int* dstp = ei + E;   // edge_index[1]

    float* ws = (float*)d_ws;
    size_t nf = (size_t)N * NDIM;
    float* x0    = ws; ws += nf;
    float* x1    = ws; ws += nf;
    float* P     = ws; ws += 6 * nf;   // Qr,Kr,Vr,Qc,Kc,Vc
    float* agg_r = ws; ws += nf;
    float* agg_c = ws; ws += nf;
    float* scr   = ws; ws += E;
    float* scc   = ws; ws += E;
    float* m_r   = ws; ws += N;
    float* m_c   = ws; ws += N;
    float* s_r   = ws; ws += N;
    float* s_c   = ws; ws += N;

    int nThreads = N * NDIM;
    k_gather<<<(nThreads + 255) / 256, 256, 0, stream>>>(atoms, emb, x0, N);

    float* Qr = P;          float* Kr = P + nf;     float* Vr = P + 2 * nf;
    float* Qc = P + 3 * nf; float* Kc = P + 4 * nf; float* Vc = P + 5 * nf;

    int rowBlocks = (N + 15) / 16;
    float* xcur = x0;
    for (int l = 0; l < L; ++l) {
        float* xnext = (l == L - 1) ? (float*)d_out : ((xcur == x0) ? x1 : x0);

        k_init<<<(nThreads + 255) / 256, 256, 0, stream>>>(agg_r, agg_c, m_r, m_c,
                                                            s_r, s_c, N);
        dim3 gproj(rowBlocks, 6, 1);
        k_proj6<<<gproj, 128, 0, stream>>>(
            xcur,
            Wq_r + (size_t)l * NDIM * NDIM, Wk_r + (size_t)l * NDIM * NDIM,
            Wv_r + (size_t)l * NDIM * NDIM, Wq_c + (size_t)l * NDIM * NDIM,
            Wk_c + (size_t)l * NDIM * NDIM, Wv_c + (size_t)l * NDIM * NDIM,
            P, N);

        k_edge_scores<<<((size_t)E * 4 + 255) / 256, 256, 0, stream>>>(
            srcp, dstp, eids, Qr, Kr, Qc, Kc,
            Ee_r + (size_t)l * T * NDIM, Ee_c + (size_t)l * T * NDIM,
            scr, scc, m_r, m_c, E);

        k_edge_exp<<<(E + 255) / 256, 256, 0, stream>>>(srcp, dstp, scr, scc,
                                                        m_r, m_c, s_r, s_c, E);

        k_edge_aggr<<<((size_t)E * 16 + 255) / 256, 256, 0, stream>>>(
            srcp, dstp, scr, scc, s_r, s_c, Vr, Vc, agg_r, agg_c, E);

        k_out<<<rowBlocks, 128, 0, stream>>>(xcur, agg_r, agg_c,
                                             Wa + (size_t)l * 2 * NDIM * NDIM,
                                             ba + (size_t)l * NDIM, xnext, N);
        xcur = xnext;
    }
}